// GenderClassifier_44263932953329
// MI455X (gfx1250) — compile-verified
//
#include <hip/hip_runtime.h>

typedef float v2f __attribute__((ext_vector_type(2)));
typedef float v8f __attribute__((ext_vector_type(8)));

#define B_SAMPLES   4096
#define N_EXPERTS   50
#define D_DIM       16384
#define OUT_DIM     2
#define X_ROW_STRIDE 32768   // floats per sample: [2][16384]
#define X_HALF_OFF   16384   // select x[:,1]
#define K_CHUNK      2048    // per-wave K range (8 waves * 2048 = 16384)

// ---------------- binning kernels ----------------

__global__ void k_zero(int* __restrict__ counts) {
  int i = threadIdx.x;
  if (i < N_EXPERTS) counts[i] = 0;
}

__global__ void k_index(const int* __restrict__ t, int* __restrict__ sidx,
                        int* __restrict__ counts) {
  int b = blockIdx.x * blockDim.x + threadIdx.x;
  if (b < B_SAMPLES) {
    int e = (981 - t[b]) / 20;     // TIMESTEPS[i] = 981 - 20*i  ->  index i
    sidx[b] = e;
    atomicAdd(&counts[e], 1);
  }
}

__global__ void k_scan(const int* __restrict__ counts, int* __restrict__ offsets,
                       int* __restrict__ cursor) {
  if (threadIdx.x == 0) {
    int acc = 0;
    for (int i = 0; i < N_EXPERTS; ++i) {
      offsets[i] = acc;
      cursor[i]  = acc;
      acc += counts[i];
    }
    offsets[N_EXPERTS] = acc;
  }
}

__global__ void k_scatter(const int* __restrict__ sidx, int* __restrict__ cursor,
                          int* __restrict__ binList) {
  int b = blockIdx.x * blockDim.x + threadIdx.x;
  if (b < B_SAMPLES) {
    int e = sidx[b];
    int pos = atomicAdd(&cursor[e], 1);  // order within bin irrelevant: per-sample
    binList[pos] = b;                    // result is order-independent
  }
}

// ---------------- grouped thin GEMM via f32 WMMA ----------------
// Block = (expert e, tile of 16 samples). 256 threads = 8 waves; wave w owns
// K in [w*2048, w*2048+2048). Per step k (+=4):
//   A (16x4, f32): lanes 0-15 hold row m=lane, K={k,k+1}; lanes 16-31 K={k+2,k+3}
//   B (4x16, f32): lane's column n = lane&15; lanes 0-15 hold K={k,k+1},
//                  lanes 16-31 K={k+2,k+3}. Only columns 0,1 carry W rows.
// C (16x16 f32, 8 VGPRs): lane L holds column N=L&15, rows 0-7 (L<16) / 8-15.
//
// B loads are UNCONDITIONAL (clamped to expert row 0 for lanes n>=2, L0-hot)
// and zeroed with a branchless per-lane select -> no EXEC-mask churn in the
// inner loop, just: 2x global_load_b64 + 2x v_cndmask + 1x v_wmma.

__global__ __launch_bounds__(256)
void k_expert_gemm(const float* __restrict__ x, const float* __restrict__ W,
                   const float* __restrict__ bias, const int* __restrict__ binList,
                   const int* __restrict__ counts, const int* __restrict__ offsets,
                   float* __restrict__ out) {
  const int e    = blockIdx.y;
  const int tile = blockIdx.x;
  const int cnt  = counts[e];
  if (tile * 16 >= cnt) return;          // uniform exit: EXEC stays all-1s below

  const int base = offsets[e] + tile * 16;
  int valid = cnt - tile * 16;
  if (valid > 16) valid = 16;

  __shared__ int   smp[16];
  __shared__ float part[8][16][OUT_DIM];

  const int tid = threadIdx.x;
  if (tid < 16) smp[tid] = (tid < valid) ? binList[base + tid] : -1;
  __syncthreads();

  const int wave  = tid >> 5;
  const int lane  = tid & 31;
  const int m     = lane & 15;   // A row / B column for this lane
  const int khalf = lane >> 4;   // 0: K pair {0,1}; 1: K pair {2,3}

  const int s     = smp[m];
  const int sload = (s < 0) ? smp[0] : s;          // keep all lanes loading
  const float* xrow = x + (size_t)sload * X_ROW_STRIDE + X_HALF_OFF
                        + (size_t)wave * K_CHUNK + khalf * 2;
  const bool hasW = (m < OUT_DIM);
  const float wmask = hasW ? 1.0f : 0.0f;           // loop-invariant lane mask
  const float* wrow = W + ((size_t)e * OUT_DIM + (hasW ? m : 0)) * D_DIM
                        + (size_t)wave * K_CHUNK + khalf * 2;

  v8f c = {};
#pragma unroll 8
  for (int k = 0; k < K_CHUNK; k += 4) {
    // streaming x: non-temporal so the 268MB stream doesn't evict L2-hot W
    v2f a = __builtin_nontemporal_load((const v2f*)(xrow + k));
    // unconditional W load (clamped address), branchless zero for n>=2 lanes
    v2f braw = *(const v2f*)(wrow + k);
    v2f b;
    b.x = braw.x * wmask;
    b.y = braw.y * wmask;
    // D = A(16x4) * B(4x16) + C ; args: neg_a, A, neg_b, B, c_mod, C, reuse_a, reuse_b
    c = __builtin_amdgcn_wmma_f32_16x16x4_f32(false, a, false, b,
                                              (short)0, c, false, false);
  }

  // Only columns 0/1 of C are meaningful (lanes 0,1,16,17 per wave)
  if (hasW) {
#pragma unroll
    for (int r = 0; r < 8; ++r) part[wave][khalf * 8 + r][m] = c[r];
  }
  __syncthreads();

  // cross-wave reduction + bias + scatter to output
  if (tid < 32) {
    const int mm = tid >> 1;
    const int o  = tid & 1;
    float acc = bias[e * OUT_DIM + o];
#pragma unroll
    for (int w = 0; w < 8; ++w) acc += part[w][mm][o];
    const int sm = smp[mm];
    if (sm >= 0) out[sm * OUT_DIM + o] = acc;
  }
}

// ---------------- launcher ----------------

extern "C" void kernel_launch(void* const* d_in, const int* in_sizes, int n_in,
                              void* d_out, int out_size, void* d_ws, size_t ws_size,
                              hipStream_t stream) {
  (void)in_sizes; (void)n_in; (void)out_size; (void)ws_size;

  const float* x    = (const float*)d_in[0];   // [B,2,4,64,64] f32
  const int*   t    = (const int*)  d_in[1];   // [B] i32
  const float* W    = (const float*)d_in[2];   // [50,2,16384] f32
  const float* bias = (const float*)d_in[3];   // [50,2] f32
  float*       out  = (float*)d_out;           // [B,2] f32

  int* ws      = (int*)d_ws;
  int* sidx    = ws;                      // 4096 ints
  int* binList = ws + B_SAMPLES;          // 4096 ints
  int* counts  = ws + 2 * B_SAMPLES;      // 50 ints
  int* offsets = counts + 64;             // 51 ints
  int* cursor  = offsets + 64;            // 50 ints   (~33 KB total)

  hipLaunchKernelGGL(k_zero,    dim3(1),  dim3(64),  0, stream, counts);
  hipLaunchKernelGGL(k_index,   dim3(B_SAMPLES / 256), dim3(256), 0, stream,
                     t, sidx, counts);
  hipLaunchKernelGGL(k_scan,    dim3(1),  dim3(32),  0, stream,
                     counts, offsets, cursor);
  hipLaunchKernelGGL(k_scatter, dim3(B_SAMPLES / 256), dim3(256), 0, stream,
                     sidx, cursor, binList);

  dim3 grid(B_SAMPLES / 16, N_EXPERTS);   // worst-case tiles; empty tiles exit
  hipLaunchKernelGGL(k_expert_gemm, grid, dim3(256), 0, stream,
                     x, W, bias, binList, counts, offsets, out);
}